// GGNN_5085241279192
// MI455X (gfx1250) — compile-verified
//
#include <hip/hip_runtime.h>
#include <hip/hip_bf16.h>

// GGNN on MI455X (gfx1250, wave32). Edge MLP via V_WMMA_F32_16X16X4_F32
// (exact f32 matrix path). Edge tensors are never materialized.

typedef __attribute__((ext_vector_type(2))) float v2f;
typedef __attribute__((ext_vector_type(8))) float v8f;

#define N_NODES 512
#define S_DIM 16
#define M_DIM 32
#define HM 64
#define HR 64
#define W1_COLS 35   // 2*S + 3

__device__ __forceinline__ v8f wmma4(v2f a, v2f b, v8f c) {
    // D = A(16x4,f32) * B(4x16,f32) + C(16x16,f32)
    return __builtin_amdgcn_wmma_f32_16x16x4_f32(
        /*neg_a=*/false, a, /*neg_b=*/false, b,
        /*c_mod=*/(short)0, c, /*reuse_a=*/false, /*reuse_b=*/false);
}

__device__ __forceinline__ float sigmoidf_(float x) {
    return 1.0f / (1.0f + __expf(-x));
}

// ---------------------------------------------------------------- init h = 0
__global__ void ggnn_init_h(float* __restrict__ h) {
    for (int i = threadIdx.x; i < N_NODES * S_DIM; i += blockDim.x) h[i] = 0.0f;
}

// ---------------------------------------- per-step node projections + zero m
// A[i,k] = sum_s h[i,s]*W1[k,s]        + b[i]*W1[k,33] + b1[k]
// B[j,k] = sum_s h[j,s]*W1[k,16+s]     + b[j]*W1[k,34]
__global__ void ggnn_node_pre(const float* __restrict__ h,
                              const float* __restrict__ b,
                              const float* __restrict__ W1,
                              const float* __restrict__ b1,
                              float* __restrict__ A,
                              float* __restrict__ B,
                              float* __restrict__ m) {
    int t = blockIdx.x * blockDim.x + threadIdx.x;    // 128*256 = 32768
    int i = t >> 6;
    int k = t & 63;
    const float* w = &W1[k * W1_COLS];
    const float* hi = &h[i * S_DIM];
    float a = b1[k] + b[i] * w[33];
    float bb = b[i] * w[34];
#pragma unroll
    for (int s = 0; s < S_DIM; ++s) {
        float hv = hi[s];
        a  += hv * w[s];
        bb += hv * w[16 + s];
    }
    A[t] = a;
    B[t] = bb;
    if (t < N_NODES * M_DIM) m[t] = 0.0f;             // zero the message accumulator
}

// ------------------------------------------------------------- edge kernel
// grid: (32 j-tiles, 8 i-chunks), block: 256 threads = 8 waves.
// Wave w handles j-tile [j0, j0+16) and 8 source nodes i.
__global__ void __launch_bounds__(256)
ggnn_edge(const float* __restrict__ J,
          const float* __restrict__ A,
          const float* __restrict__ B,
          const float* __restrict__ W1,
          const float* __restrict__ b2,
          const float* __restrict__ W2,
          const float* __restrict__ W3,
          float* __restrict__ m) {
    __shared__ float sB[16 * 64];       // B-tile for these 16 j's
    __shared__ float sW2[64 * 64];      // W2 row-major (o,k)
    __shared__ float sW3[32 * 64];      // W3 row-major (o,k)
    __shared__ float swJ[64];
    __shared__ float sb2[64];
    __shared__ float sx[8 * 1024];      // per-wave 16x64 scratch (x1, then x2)

    const int tid  = threadIdx.x;
    const int j0   = blockIdx.x * 16;
    const int i0   = blockIdx.y * 64;
    const int lane = tid & 31;
    const int wave = tid >> 5;

    for (int x = tid; x < 16 * 64; x += 256) sB[x] = B[j0 * 64 + x];
    for (int x = tid; x < 64 * 64; x += 256) sW2[x] = W2[x];
    for (int x = tid; x < 32 * 64; x += 256) sW3[x] = W3[x];
    if (tid < 64) { swJ[tid] = W1[tid * W1_COLS + 32]; sb2[tid] = b2[tid]; }
    __syncthreads();

    float* xs = &sx[wave << 10];                 // wave-private 16x64
    const int lm   = lane & 15;
    const int half = (lane < 16) ? 0 : 2;        // K sub-offset per ISA A/B layout

    v8f macc0 = {0,0,0,0,0,0,0,0};               // msg cols 0..15  (C/D layout)
    v8f macc1 = {0,0,0,0,0,0,0,0};               // msg cols 16..31

    for (int it = 0; it < 8; ++it) {
        const int i = i0 + wave * 8 + it;
        const float* Ai = &A[i * 64];
        const float* Ji = &J[i * N_NODES + j0];

        // x1[j',k] = relu(A[i,k] + B[j0+j',k] + J[i,j0+j']*wJ[k])
#pragma unroll
        for (int r = 0; r < 32; ++r) {
            int idx = lane + r * 32;
            int jj = idx >> 6, k = idx & 63;
            float v = Ai[k] + sB[idx] + Ji[jj] * swJ[k];
            xs[idx] = fmaxf(v, 0.0f);
        }

        // stage 1: X2 = relu(X1 @ W2^T + b2)   (4 output tiles, K=64)
        v8f c0, c1, c2, c3;
        {
            float bz;
            bz = sb2[lm];      c0 = (v8f){bz,bz,bz,bz,bz,bz,bz,bz};
            bz = sb2[16 + lm]; c1 = (v8f){bz,bz,bz,bz,bz,bz,bz,bz};
            bz = sb2[32 + lm]; c2 = (v8f){bz,bz,bz,bz,bz,bz,bz,bz};
            bz = sb2[48 + lm]; c3 = (v8f){bz,bz,bz,bz,bz,bz,bz,bz};
        }
#pragma unroll
        for (int t = 0; t < 16; ++t) {
            v2f a = *(const v2f*)&xs[lm * 64 + 4 * t + half];
            v2f w0 = *(const v2f*)&sW2[(0  + lm) * 64 + 4 * t + half];
            v2f w1 = *(const v2f*)&sW2[(16 + lm) * 64 + 4 * t + half];
            v2f w2 = *(const v2f*)&sW2[(32 + lm) * 64 + 4 * t + half];
            v2f w3 = *(const v2f*)&sW2[(48 + lm) * 64 + 4 * t + half];
            c0 = wmma4(a, w0, c0);
            c1 = wmma4(a, w1, c1);
            c2 = wmma4(a, w2, c2);
            c3 = wmma4(a, w3, c3);
        }
        // relu + write X2 back (D layout -> row-major LDS), overwriting x1
        const int rbase = (lane < 16) ? 0 : 8;
#pragma unroll
        for (int v = 0; v < 8; ++v) {
            int row = (v + rbase) * 64;
            xs[row + 0  + lm] = fmaxf(c0[v], 0.0f);
            xs[row + 16 + lm] = fmaxf(c1[v], 0.0f);
            xs[row + 32 + lm] = fmaxf(c2[v], 0.0f);
            xs[row + 48 + lm] = fmaxf(c3[v], 0.0f);
        }

        // stage 2: macc += X2 @ W3^T   (accumulate message over i in-register)
#pragma unroll
        for (int t = 0; t < 16; ++t) {
            v2f a  = *(const v2f*)&xs[lm * 64 + 4 * t + half];
            v2f w0 = *(const v2f*)&sW3[(0  + lm) * 64 + 4 * t + half];
            v2f w1 = *(const v2f*)&sW3[(16 + lm) * 64 + 4 * t + half];
            macc0 = wmma4(a, w0, macc0);
            macc1 = wmma4(a, w1, macc1);
        }
    }

    // flush per-wave partial sums: element (M=j', N=col) -> m[j0+j', col]
    const int rbase = (lane < 16) ? 0 : 8;
#pragma unroll
    for (int v = 0; v < 8; ++v) {
        int j = j0 + v + rbase;
        atomicAdd(&m[j * M_DIM + 0  + lm], macc0[v]);
        atomicAdd(&m[j * M_DIM + 16 + lm], macc1[v]);
    }
}

// ---------------------------------------------------------------- GRU update
__global__ void ggnn_gru(float* __restrict__ h,
                         const float* __restrict__ m,
                         const float* __restrict__ b3,
                         const float* __restrict__ W_ih,
                         const float* __restrict__ b_ih,
                         const float* __restrict__ W_hh,
                         const float* __restrict__ b_hh) {
    int i = blockIdx.x * blockDim.x + threadIdx.x;   // 2*256 = 512 nodes
    float x[48];                                      // gru_in = [h, m_j]
#pragma unroll
    for (int s = 0; s < S_DIM; ++s) x[s] = h[i * S_DIM + s];
#pragma unroll
    for (int c = 0; c < M_DIM; ++c)
        x[S_DIM + c] = m[i * M_DIM + c] + 512.0f * b3[c];   // + N*b3 (per-edge bias)

    float gi[48], gh[48];
#pragma unroll
    for (int o = 0; o < 48; ++o) {
        float a = b_ih[o];
        const float* w = &W_ih[o * 48];
#pragma unroll
        for (int q = 0; q < 48; ++q) a += w[q] * x[q];
        gi[o] = a;
        float g = b_hh[o];
        const float* wh = &W_hh[o * S_DIM];
#pragma unroll
        for (int q = 0; q < S_DIM; ++q) g += wh[q] * x[q];
        gh[o] = g;
    }
#pragma unroll
    for (int s = 0; s < S_DIM; ++s) {
        float r = sigmoidf_(gi[s] + gh[s]);
        float z = sigmoidf_(gi[16 + s] + gh[16 + s]);
        float n = tanhf(gi[32 + s] + r * gh[32 + s]);
        h[i * S_DIM + s] = (1.0f - z) * n + z * x[s];
    }
}

// ----------------------------------------------------------------- readout
__global__ void ggnn_readout(const float* __restrict__ h,
                             const float* __restrict__ Wr1, const float* __restrict__ br1,
                             const float* __restrict__ Wr2, const float* __restrict__ br2,
                             const float* __restrict__ Wr3, const float* __restrict__ br3,
                             float* __restrict__ out) {
    int i = blockIdx.x * blockDim.x + threadIdx.x;   // 512 nodes
    float hv[S_DIM];
#pragma unroll
    for (int s = 0; s < S_DIM; ++s) hv[s] = h[i * S_DIM + s];

    float y1[HR];
#pragma unroll
    for (int o = 0; o < HR; ++o) {
        float a = br1[o];
        const float* w = &Wr1[o * S_DIM];
#pragma unroll
        for (int s = 0; s < S_DIM; ++s) a += w[s] * hv[s];
        y1[o] = fmaxf(a, 0.0f);
    }
    float y2[HR];
#pragma unroll
    for (int o = 0; o < HR; ++o) {
        float a = br2[o];
        const float* w = &Wr2[o * HR];
#pragma unroll
        for (int q = 0; q < HR; ++q) a += w[q] * y1[q];
        y2[o] = fmaxf(a, 0.0f);
    }
    float o0 = br3[0], o1 = br3[1];
#pragma unroll
    for (int q = 0; q < HR; ++q) {
        o0 += Wr3[q] * y2[q];
        o1 += Wr3[HR + q] * y2[q];
    }
    float s0 = sigmoidf_(o0), s1 = sigmoidf_(o1);
    float inv = 1.0f / (s0 + s1);
    out[i * 2 + 0] = s0 * inv;
    out[i * 2 + 1] = s1 * inv;
}

// ------------------------------------------------------------------- launch
extern "C" void kernel_launch(void* const* d_in, const int* in_sizes, int n_in,
                              void* d_out, int out_size, void* d_ws, size_t ws_size,
                              hipStream_t stream) {
    const float* J    = (const float*)d_in[0];
    const float* b    = (const float*)d_in[1];
    const float* W1   = (const float*)d_in[2];
    const float* b1   = (const float*)d_in[3];
    const float* W2   = (const float*)d_in[4];
    const float* b2   = (const float*)d_in[5];
    const float* W3   = (const float*)d_in[6];
    const float* b3   = (const float*)d_in[7];
    const float* W_ih = (const float*)d_in[8];
    const float* b_ih = (const float*)d_in[9];
    const float* W_hh = (const float*)d_in[10];
    const float* b_hh = (const float*)d_in[11];
    const float* Wr1  = (const float*)d_in[12];
    const float* br1  = (const float*)d_in[13];
    const float* Wr2  = (const float*)d_in[14];
    const float* br2  = (const float*)d_in[15];
    const float* Wr3  = (const float*)d_in[16];
    const float* br3  = (const float*)d_in[17];

    float* ws = (float*)d_ws;
    float* h  = ws;                       // 512*16
    float* A  = h  + N_NODES * S_DIM;     // 512*64
    float* B  = A  + N_NODES * 64;        // 512*64
    float* m  = B  + N_NODES * 64;        // 512*32

    ggnn_init_h<<<1, 256, 0, stream>>>(h);
    for (int step = 0; step < 5; ++step) {
        ggnn_node_pre<<<128, 256, 0, stream>>>(h, b, W1, b1, A, B, m);
        ggnn_edge<<<dim3(32, 8), 256, 0, stream>>>(J, A, B, W1, b2, W2, W3, m);
        ggnn_gru<<<2, 256, 0, stream>>>(h, m, b3, W_ih, b_ih, W_hh, b_hh);
    }
    ggnn_readout<<<2, 256, 0, stream>>>(h, Wr1, br1, Wr2, br2, Wr3, br3,
                                        (float*)d_out);
}